// LlamaApproximatedAttention_87093346828498
// MI455X (gfx1250) — compile-verified
//
#include <hip/hip_runtime.h>

// CDNA5 / gfx1250, wave32. fp32 WMMA: V_WMMA_F32_16X16X4_F32
//   A: 16x4 f32 -> v2f/lane, B: 4x16 f32 -> v2f/lane, C/D: 16x16 -> v8f/lane.
// out[b] = (hid[b] @ W_hid^T) @ (W_seq @ hid[b]), rank R=64.

typedef __attribute__((ext_vector_type(2))) float v2f;
typedef __attribute__((ext_vector_type(4))) float v4f;
typedef __attribute__((ext_vector_type(8))) float v8f;

static __device__ __forceinline__ v8f wmma4(v2f a, v2f b, v8f c) {
  // 8 args: (neg_a, A, neg_b, B, c_mod, C, reuse_a, reuse_b)
  return __builtin_amdgcn_wmma_f32_16x16x4_f32(false, a, false, b, (short)0, c,
                                               false, false);
}

// ---------------------------------------------------------------------------
// NT kernel (B stored [N,K]): per-wave 32x64 tile, register double-buffered.
// Used for Fs = hid @ W_hid^T (B = W_hid is 512 KB, L2-resident, contiguous
// v2f fragment loads -> no LDS staging needed).
// ---------------------------------------------------------------------------
__global__ __launch_bounds__(256) void gemm_nt_wmma(
    const float* __restrict__ A, const float* __restrict__ Bt,
    float* __restrict__ C, int K, int lda, int ldb, int ldc, long sA, long sB,
    long sC, int tilesM, int tilesN, int nWaves) {
  int gid = (int)((blockIdx.x * blockDim.x + threadIdx.x) >> 5);
  if (gid >= nWaves) return;  // uniform per wave
  int perBatch = tilesM * tilesN;
  int b = gid / perBatch;
  int t = gid - b * perBatch;
  int mt = t / tilesN;
  int nt = t - mt * tilesN;

  int lane = (int)(threadIdx.x & 31);
  int lo = lane & 15;
  int kb = (lane >> 4) << 1;
  int half = lane >> 4;

  const float* Ab0 = A + (long)b * sA + (long)(mt * 32 + lo) * lda + kb;
  const float* Ab1 = Ab0 + (long)16 * lda;
  const float* Bb = Bt + (long)b * sB + (long)(nt * 64 + lo) * ldb + kb;
  float* Cb = C + (long)b * sC;

  v8f acc[8];
#pragma unroll
  for (int j = 0; j < 8; ++j) acc[j] = (v8f)(0.0f);

  v2f a0c = *(const v2f*)(Ab0);
  v2f a1c = *(const v2f*)(Ab1);
  v2f bc[4];
#pragma unroll
  for (int j = 0; j < 4; ++j) bc[j] = *(const v2f*)(Bb + (long)(j * 16) * ldb);

#pragma unroll 4
  for (int k = 0; k < K - 4; k += 4) {
    v2f a0n = *(const v2f*)(Ab0 + k + 4);
    v2f a1n = *(const v2f*)(Ab1 + k + 4);
    v2f bn[4];
#pragma unroll
    for (int j = 0; j < 4; ++j)
      bn[j] = *(const v2f*)(Bb + (long)(j * 16) * ldb + k + 4);

#pragma unroll
    for (int j = 0; j < 4; ++j) {
      acc[j] = wmma4(a0c, bc[j], acc[j]);
      acc[4 + j] = wmma4(a1c, bc[j], acc[4 + j]);
    }
    a0c = a0n;
    a1c = a1n;
#pragma unroll
    for (int j = 0; j < 4; ++j) bc[j] = bn[j];
  }
#pragma unroll
  for (int j = 0; j < 4; ++j) {
    acc[j] = wmma4(a0c, bc[j], acc[j]);
    acc[4 + j] = wmma4(a1c, bc[j], acc[4 + j]);
  }

#pragma unroll
  for (int mi = 0; mi < 2; ++mi)
#pragma unroll
    for (int j = 0; j < 4; ++j)
#pragma unroll
      for (int i = 0; i < 8; ++i) {
        int m = mt * 32 + mi * 16 + half * 8 + i;
        Cb[(long)m * ldc + nt * 64 + j * 16 + lo] = acc[mi * 4 + j][i];
      }
}

// ---------------------------------------------------------------------------
// NN kernel (B stored [K,N]) with cooperative LDS staging of the B panel.
// A block of W waves computes (W*32) x 64 of C; all waves share one 64-wide
// B panel, double-buffered in LDS in KC-deep chunks. Global->LDS loads of
// chunk i+1 overlap the WMMAs of chunk i. Row pad 72 floats: the two lane
// halves of a ds_load hit disjoint bank sets (2*72 mod 64 = 16).
// ---------------------------------------------------------------------------
#define KC 32
#define LDS_PITCH 72

__global__ __launch_bounds__(256) void gemm_nn_lds_wmma(
    const float* __restrict__ A, const float* __restrict__ Bm,
    float* __restrict__ C, int K, int lda, int ldb, int ldc, long sA, long sB,
    long sC, int mBlocks, int tilesN) {
  __shared__ float Bs[2][KC * LDS_PITCH];

  int perBatch = mBlocks * tilesN;
  int b = (int)blockIdx.x / perBatch;
  int t = (int)blockIdx.x - b * perBatch;
  int mb = t / tilesN;
  int nt = t - mb * tilesN;

  int tid = (int)threadIdx.x;
  int lane = tid & 31;
  int waveId = tid >> 5;
  int lo = lane & 15;
  int kb = (lane >> 4) << 1;
  int half = lane >> 4;

  int m0 = mb * (int)blockDim.x + waveId * 32;  // 32 rows per wave

  const float* Ab0 = A + (long)b * sA + (long)(m0 + lo) * lda + kb;
  const float* Ab1 = Ab0 + (long)16 * lda;
  const float* Bpanel = Bm + (long)b * sB + nt * 64;  // B[k][n0..n0+63]
  float* Cb = C + (long)b * sC;

  // cooperative stage: KC x 64 floats, coalesced float4 per thread
  auto stage = [&](int buf, int kc0) {
#pragma unroll 2
    for (int f = tid; f < KC * 16; f += (int)blockDim.x) {
      int kk = f >> 4;
      int n4 = (f & 15) << 2;
      v4f v = *(const v4f*)(Bpanel + (long)(kc0 + kk) * ldb + n4);
      *(v4f*)(&Bs[buf][kk * LDS_PITCH + n4]) = v;
    }
  };

  v8f acc[8];
#pragma unroll
  for (int j = 0; j < 8; ++j) acc[j] = (v8f)(0.0f);

  stage(0, 0);
  __syncthreads();

  int buf = 0;
  for (int kc0 = 0; kc0 < K; kc0 += KC) {
    if (kc0 + KC < K) stage(buf ^ 1, kc0 + KC);  // prefetch next chunk

    const float* Bsb = &Bs[buf][0];
#pragma unroll
    for (int kk = 0; kk < KC; kk += 4) {
      v2f a0 = *(const v2f*)(Ab0 + kc0 + kk);
      v2f a1 = *(const v2f*)(Ab1 + kc0 + kk);
#pragma unroll
      for (int j = 0; j < 4; ++j) {
        v2f bf;
        bf.x = Bsb[(kk + kb) * LDS_PITCH + j * 16 + lo];
        bf.y = Bsb[(kk + kb + 1) * LDS_PITCH + j * 16 + lo];
        acc[j] = wmma4(a0, bf, acc[j]);
        acc[4 + j] = wmma4(a1, bf, acc[4 + j]);
      }
    }
    __syncthreads();
    buf ^= 1;
  }

#pragma unroll
  for (int mi = 0; mi < 2; ++mi)
#pragma unroll
    for (int j = 0; j < 4; ++j)
#pragma unroll
      for (int i = 0; i < 8; ++i) {
        int m = m0 + mi * 16 + half * 8 + i;
        Cb[(long)m * ldc + nt * 64 + j * 16 + lo] = acc[mi * 4 + j][i];
      }
}

// ---------------------------------------------------------------------------
//   Fs[b] (S x R) = hid[b] (S x H) @ W_hid (R x H)^T   -> gemm_nt
//   G [b] (R x H) = W_seq (R x S)  @ hid[b] (S x H)    -> gemm_nn_lds
//   out[b] (S x H) = Fs[b] (S x R) @ G[b] (R x H)      -> gemm_nn_lds
// ---------------------------------------------------------------------------
extern "C" void kernel_launch(void* const* d_in, const int* in_sizes, int n_in,
                              void* d_out, int out_size, void* d_ws,
                              size_t ws_size, hipStream_t stream) {
  (void)in_sizes; (void)n_in; (void)out_size; (void)ws_size;

  constexpr int Bn = 4, S = 2048, H = 2048, R = 64;

  const float* hid   = (const float*)d_in[0];
  // d_in[1] = grid_chw (int64 index grid) — dead input for the fused GEMM form
  const float* W_seq = (const float*)d_in[2];  // [R, S]
  const float* W_hid = (const float*)d_in[3];  // [R, H]
  float* out = (float*)d_out;                  // [Bn, S, H]

  float* Fs = (float*)d_ws;                    // [Bn, S, R]  2 MB
  float* G  = Fs + (size_t)Bn * S * R;         // [Bn, R, H]  2 MB

  // 1) Fs[b] = hid[b] @ W_hid^T : M=S, N=R, K=H   (nt)
  {
    int tilesM = S / 32, tilesN = R / 64;  // 64, 1
    int nWaves = Bn * tilesM * tilesN;     // 256
    int blocks = (nWaves * 32) / 256;      // 32
    gemm_nt_wmma<<<blocks, 256, 0, stream>>>(
        hid, W_hid, Fs, /*K=*/H, /*lda=*/H, /*ldb=*/H, /*ldc=*/R,
        /*sA=*/(long)S * H, /*sB=*/0L, /*sC=*/(long)S * R, tilesM, tilesN,
        nWaves);
  }

  // 2) G[b] = W_seq @ hid[b] : M=R(=64), N=H, K=S   (nn; 2 waves/block)
  {
    int mBlocks = R / 64, tilesN = H / 64;  // 1, 32
    int blocks = Bn * mBlocks * tilesN;     // 128
    gemm_nn_lds_wmma<<<blocks, 64, 0, stream>>>(
        W_seq, hid, G, /*K=*/S, /*lda=*/S, /*ldb=*/H, /*ldc=*/H,
        /*sA=*/0L, /*sB=*/(long)S * H, /*sC=*/(long)R * H, mBlocks, tilesN);
  }

  // 3) out[b] = Fs[b] @ G[b] : M=S, N=H, K=R   (nn; 8 waves/block)
  {
    int mBlocks = S / 256, tilesN = H / 64;  // 8, 32
    int blocks = Bn * mBlocks * tilesN;      // 1024
    gemm_nn_lds_wmma<<<blocks, 256, 0, stream>>>(
        Fs, G, out, /*K=*/R, /*lda=*/R, /*ldb=*/H, /*ldc=*/H,
        /*sA=*/(long)S * R, /*sB=*/(long)R * H, /*sC=*/(long)S * H, mBlocks,
        tilesN);
  }
}